// SelectiveSSM_6923487282737
// MI455X (gfx1250) — compile-verified
//
#include <hip/hip_runtime.h>
#include <hip/hip_bf16.h>

typedef __attribute__((ext_vector_type(2))) float v2f;
typedef __attribute__((ext_vector_type(8))) float v8f;

#define D_INNER 1024
#define D_STATE 16
#define T_LEN   2048
#define BATCH   2
#define M_ROWS  (BATCH * T_LEN)     // 4096 flattened rows
#define CHUNK   128
#define NCHUNK  (T_LEN / CHUNK)     // 16 chunks per sequence

// ---------------------------------------------------------------------------
// Kernel 1: fused projections via V_WMMA_F32_16X16X4_F32.
//   dt   = softplus(u @ W_dt^T + b_dt)   (4096 x 1024)
//   Bssm = u @ W_B^T                     (4096 x 16)
//   Cssm = u @ W_C^T                     (4096 x 16)
// One block per 16-row M tile; u tile staged in LDS (64 KB). Each wave owns
// TWO adjacent output column tiles -> two independent WMMA accumulator
// chains (no serial WMMA->WMMA dependency), shared A-fragment loads.
// W rows are streamed with software prefetch (global_prefetch_b8).
// ---------------------------------------------------------------------------
__global__ __launch_bounds__(256) void ssm_proj_wmma_kernel(
    const float* __restrict__ u, const float* __restrict__ W_B,
    const float* __restrict__ W_C, const float* __restrict__ W_dt,
    const float* __restrict__ b_dt,
    float* __restrict__ dt_out, float* __restrict__ Bssm,
    float* __restrict__ Cssm)
{
    __shared__ float lds_u[16 * D_INNER];

    const int m_base = blockIdx.x * 16;
    const int tid    = threadIdx.x;

    // 16 rows of u are contiguous in flattened (B*T, D): one linear 64KB copy.
    {
        const float4* src = (const float4*)(u + (size_t)m_base * D_INNER);
        float4*       dst = (float4*)lds_u;
        #pragma unroll 4
        for (int i = tid; i < (16 * D_INNER) / 4; i += 256) dst[i] = src[i];
    }
    __syncthreads();

    const int wave = tid >> 5;
    const int lane = tid & 31;
    const int half = lane >> 4;   // 0: K pair {0,1}; 1: K pair {2,3}
    const int lr   = lane & 15;   // M index for A, N index for B

    const float* arow = lds_u + lr * D_INNER + 2 * half;

    // --- dt tiles: 32 pairs of adjacent 16-col tiles, dual accumulators ---
    for (int jp = wave; jp < 32; jp += 8) {
        const int colb0 = (2 * jp) * 16;
        const int colb1 = colb0 + 16;
        const float* brow0 = W_dt + (size_t)(colb0 + lr) * D_INNER + 2 * half;
        const float* brow1 = W_dt + (size_t)(colb1 + lr) * D_INNER + 2 * half;

        v8f c0 = {}, c1 = {};
        for (int k0 = 0; k0 < D_INNER; k0 += 64) {
            // stream-ahead prefetch of next 64-K slab of both W rows
            __builtin_prefetch(brow0 + k0 + 64, 0, 1);
            __builtin_prefetch(brow1 + k0 + 64, 0, 1);
            #pragma unroll
            for (int kk = 0; kk < 64; kk += 4) {
                v2f a  = *(const v2f*)(arow  + k0 + kk);
                v2f b0 = *(const v2f*)(brow0 + k0 + kk);
                v2f b1 = *(const v2f*)(brow1 + k0 + kk);
                c0 = __builtin_amdgcn_wmma_f32_16x16x4_f32(
                         false, a, false, b0, (short)0, c0, false, false);
                c1 = __builtin_amdgcn_wmma_f32_16x16x4_f32(
                         false, a, false, b1, (short)0, c1, false, false);
            }
        }

        const float bias0 = b_dt[colb0 + lr];
        const float bias1 = b_dt[colb1 + lr];
        #pragma unroll
        for (int r = 0; r < 8; ++r) {
            const int m = m_base + r + 8 * half;
            float x0 = c0[r] + bias0;
            float x1 = c1[r] + bias1;
            // stable softplus = max(x,0) + log1p(exp(-|x|))
            dt_out[(size_t)m * D_INNER + colb0 + lr] =
                fmaxf(x0, 0.0f) + log1pf(expf(-fabsf(x0)));
            dt_out[(size_t)m * D_INNER + colb1 + lr] =
                fmaxf(x1, 0.0f) + log1pf(expf(-fabsf(x1)));
        }
    }

    // --- B_ssm (wave 0) and C_ssm (wave 1): single 16-col tile each ---
    if (wave < 2) {
        const float* W   = (wave == 0) ? W_B : W_C;
        float*       out = (wave == 0) ? Bssm : Cssm;
        const float* brow = W + (size_t)lr * D_INNER + 2 * half;

        v8f c = {};
        for (int k0 = 0; k0 < D_INNER; k0 += 64) {
            __builtin_prefetch(brow + k0 + 64, 0, 1);
            #pragma unroll
            for (int kk = 0; kk < 64; kk += 4) {
                v2f a = *(const v2f*)(arow + k0 + kk);
                v2f b = *(const v2f*)(brow + k0 + kk);
                c = __builtin_amdgcn_wmma_f32_16x16x4_f32(
                        false, a, false, b, (short)0, c, false, false);
            }
        }
        #pragma unroll
        for (int r = 0; r < 8; ++r) {
            const int m = m_base + r + 8 * half;
            out[(size_t)m * D_STATE + lr] = c[r];
        }
    }
}

__device__ inline void load16f(const float* __restrict__ p, float* out)
{
    const float4* q = (const float4*)p;
    float4 a = q[0], b = q[1], cc = q[2], d = q[3];
    out[0] = a.x;  out[1] = a.y;  out[2] = a.z;  out[3] = a.w;
    out[4] = b.x;  out[5] = b.y;  out[6] = b.z;  out[7] = b.w;
    out[8] = cc.x; out[9] = cc.y; out[10] = cc.z; out[11] = cc.w;
    out[12] = d.x; out[13] = d.y; out[14] = d.z; out[15] = d.w;
}

// ---------------------------------------------------------------------------
// Kernel 2 (scan phase A): per (b, chunk g, channel d) compute the chunk's
// composed affine transform per state n: P = prod(dA_t), Q = local scan from 0.
// dA, dBu recomputed on the fly (never materialized: saves ~0.5 GB traffic).
// 32768 threads -> good occupancy despite serial T inside a chunk.
// ---------------------------------------------------------------------------
__global__ __launch_bounds__(256) void ssm_chunk_scan_kernel(
    const float* __restrict__ u, const float* __restrict__ dt_ws,
    const float* __restrict__ Bssm, const float* __restrict__ log_A,
    float* __restrict__ Pws, float* __restrict__ Qws)
{
    const int idx = blockIdx.x * 256 + threadIdx.x;   // ((b*G + g)*D + d)
    const int d   = idx & (D_INNER - 1);
    const int bg  = idx / D_INNER;
    const int g   = bg & (NCHUNK - 1);
    const int b   = bg / NCHUNK;

    float Av[D_STATE];
    #pragma unroll
    for (int n = 0; n < D_STATE; ++n) Av[n] = -expf(log_A[d * D_STATE + n]);

    float P[D_STATE], Q[D_STATE];
    #pragma unroll
    for (int n = 0; n < D_STATE; ++n) { P[n] = 1.0f; Q[n] = 0.0f; }

    const int t0 = g * CHUNK;
    for (int t = t0; t < t0 + CHUNK; ++t) {
        const size_t row = (size_t)(b * T_LEN + t);
        const float dtv = dt_ws[row * D_INNER + d];
        const float uv  = u[row * D_INNER + d];
        float Bs[D_STATE];
        load16f(Bssm + row * D_STATE, Bs);
        const float dtu = dtv * uv;
        #pragma unroll
        for (int n = 0; n < D_STATE; ++n) {
            const float a = expf(dtv * Av[n]);   // arg <= 0, a in (0,1]
            P[n] *= a;
            Q[n]  = fmaf(a, Q[n], dtu * Bs[n]);
        }
    }

    const size_t base = ((size_t)(b * D_INNER + d) * D_STATE) * NCHUNK + g;
    #pragma unroll
    for (int n = 0; n < D_STATE; ++n) {
        Pws[base + (size_t)n * NCHUNK] = P[n];
        Qws[base + (size_t)n * NCHUNK] = Q[n];
    }
}

// ---------------------------------------------------------------------------
// Kernel 3 (scan phase B): exclusive combine over the 16 chunks per (b,d,n).
// Tiny: 2048 threads, 16 serial steps.
// ---------------------------------------------------------------------------
__global__ __launch_bounds__(256) void ssm_combine_kernel(
    const float* __restrict__ Pws, const float* __restrict__ Qws,
    float* __restrict__ Xinit)
{
    const int idx = blockIdx.x * 256 + threadIdx.x;   // b*D + d
    const size_t base = (size_t)idx * D_STATE * NCHUNK;

    float x[D_STATE];
    #pragma unroll
    for (int n = 0; n < D_STATE; ++n) x[n] = 0.0f;

    for (int g = 0; g < NCHUNK; ++g) {
        #pragma unroll
        for (int n = 0; n < D_STATE; ++n) {
            const size_t o = base + (size_t)n * NCHUNK + g;
            Xinit[o] = x[n];                       // exclusive prefix state
            x[n] = fmaf(Pws[o], x[n], Qws[o]);
        }
    }
}

// ---------------------------------------------------------------------------
// Kernel 4 (scan phase C): re-scan each chunk from its true initial state,
// emit y[b,t,d] = sum_n x*C + u*D_param. d-contiguous thread map -> coalesced.
// ---------------------------------------------------------------------------
__global__ __launch_bounds__(256) void ssm_output_kernel(
    const float* __restrict__ u, const float* __restrict__ dt_ws,
    const float* __restrict__ Bssm, const float* __restrict__ Cssm,
    const float* __restrict__ log_A, const float* __restrict__ D_param,
    const float* __restrict__ Xinit, float* __restrict__ y)
{
    const int idx = blockIdx.x * 256 + threadIdx.x;
    const int d   = idx & (D_INNER - 1);
    const int bg  = idx / D_INNER;
    const int g   = bg & (NCHUNK - 1);
    const int b   = bg / NCHUNK;

    float Av[D_STATE];
    #pragma unroll
    for (int n = 0; n < D_STATE; ++n) Av[n] = -expf(log_A[d * D_STATE + n]);

    const size_t base = ((size_t)(b * D_INNER + d) * D_STATE) * NCHUNK + g;
    float x[D_STATE];
    #pragma unroll
    for (int n = 0; n < D_STATE; ++n) x[n] = Xinit[base + (size_t)n * NCHUNK];

    const float Dp = D_param[d];

    const int t0 = g * CHUNK;
    for (int t = t0; t < t0 + CHUNK; ++t) {
        const size_t row = (size_t)(b * T_LEN + t);
        const float dtv = dt_ws[row * D_INNER + d];
        const float uv  = u[row * D_INNER + d];
        float Bs[D_STATE], Cs[D_STATE];
        load16f(Bssm + row * D_STATE, Bs);
        load16f(Cssm + row * D_STATE, Cs);
        const float dtu = dtv * uv;
        float acc = 0.0f;
        #pragma unroll
        for (int n = 0; n < D_STATE; ++n) {
            const float a = expf(dtv * Av[n]);
            x[n] = fmaf(a, x[n], dtu * Bs[n]);
            acc  = fmaf(x[n], Cs[n], acc);
        }
        y[row * D_INNER + d] = fmaf(uv, Dp, acc);
    }
}

extern "C" void kernel_launch(void* const* d_in, const int* in_sizes, int n_in,
                              void* d_out, int out_size, void* d_ws, size_t ws_size,
                              hipStream_t stream)
{
    const float* u       = (const float*)d_in[0];
    const float* W_B     = (const float*)d_in[1];
    const float* W_C     = (const float*)d_in[2];
    const float* W_dt    = (const float*)d_in[3];
    const float* b_dt    = (const float*)d_in[4];
    const float* log_A   = (const float*)d_in[5];
    const float* D_param = (const float*)d_in[6];
    float* y = (float*)d_out;

    // Workspace layout (floats), all offsets 64B aligned:
    float* ws      = (float*)d_ws;
    float* dt_ws   = ws;                                   // 4096*1024
    float* Bssm    = dt_ws + (size_t)M_ROWS * D_INNER;     // 4096*16
    float* Cssm    = Bssm  + (size_t)M_ROWS * D_STATE;     // 4096*16
    float* Pws     = Cssm  + (size_t)M_ROWS * D_STATE;     // 2*1024*16*16
    float* Qws     = Pws   + (size_t)BATCH * D_INNER * D_STATE * NCHUNK;
    float* Xinit   = Qws   + (size_t)BATCH * D_INNER * D_STATE * NCHUNK;

    // 1) Projections (WMMA f32): 256 M-tiles, 8 waves each, 2 tiles/wave.
    ssm_proj_wmma_kernel<<<M_ROWS / 16, 256, 0, stream>>>(
        u, W_B, W_C, W_dt, b_dt, dt_ws, Bssm, Cssm);

    // 2) Per-chunk scan: (B * NCHUNK * D_INNER) threads.
    ssm_chunk_scan_kernel<<<(BATCH * NCHUNK * D_INNER) / 256, 256, 0, stream>>>(
        u, dt_ws, Bssm, log_A, Pws, Qws);

    // 3) Combine chunk transforms (exclusive): B*D threads.
    ssm_combine_kernel<<<(BATCH * D_INNER) / 256, 256, 0, stream>>>(
        Pws, Qws, Xinit);

    // 4) Final chunk re-scan + output.
    ssm_output_kernel<<<(BATCH * NCHUNK * D_INNER) / 256, 256, 0, stream>>>(
        u, dt_ws, Bssm, Cssm, log_A, D_param, Xinit, y);

    (void)in_sizes; (void)n_in; (void)out_size; (void)ws_size;
}